// KernelAMController_88708254532320
// MI455X (gfx1250) — compile-verified
//
#include <hip/hip_runtime.h>
#include <math.h>

// ---------------------------------------------------------------------------
// Gaussian-KDE weighted adjoint gather, binned by time, WMMA-accelerated.
//   out[b,:] = - (sum_g w_eff[b,g] * adj[tb_b,g,:]) / (sum_g w_eff[b,g] + eps)
//   w_eff    = exp(-2*||x_b - p_g||^2) * (counts[tb_b,g] > 0)
// Inner loop: V_WMMA_F32_16X16X4_F32 produces 256 log2-weights per issue.
// ---------------------------------------------------------------------------

typedef __attribute__((ext_vector_type(2))) float v2f;
typedef __attribute__((ext_vector_type(8))) float v8f;

#define WAVES_PER_BLOCK 4
#define CTHREADS (WAVES_PER_BLOCK * 32)
#define C2LOG2E 2.885390081777927f   // 2*log2(e)

__device__ __forceinline__ float fast_exp2(float x) {
#if __has_builtin(__builtin_amdgcn_exp2f)
    return __builtin_amdgcn_exp2f(x);   // raw v_exp_f32
#else
    return exp2f(x);
#endif
}

// ---- pass 0: zero the counters region of the workspace ---------------------
__global__ void k_init(int* w) {
    if (threadIdx.x < 96) w[threadIdx.x] = 0;
}

// ---- pass 1: per-sample time bin + histogram -------------------------------
__global__ void k_bin(const float* __restrict__ t, const float* __restrict__ edges,
                      int* __restrict__ tb, int* __restrict__ cnts, int B, int NB) {
    int i = blockIdx.x * blockDim.x + threadIdx.x;
    if (i >= B) return;
    float tv = t[i];
    int b = 0;
    for (int j = 1; j <= NB - 1; ++j) b += (edges[j] < tv) ? 1 : 0;
    if (b > NB - 1) b = NB - 1;
    tb[i] = b;
    atomicAdd(&cnts[b], 1);
}

// ---- pass 2: exclusive scan of bin counts ----------------------------------
__global__ void k_scan(const int* __restrict__ cnts, int* __restrict__ offs,
                       int* __restrict__ cur, int NB) {
    if (threadIdx.x == 0) {
        int run = 0;
        for (int b = 0; b < NB; ++b) { offs[b] = run; cur[b] = run; run += cnts[b]; }
        offs[NB] = run;
    }
}

// ---- pass 3: scatter sample ids into bin-sorted order ----------------------
__global__ void k_scatter(const int* __restrict__ tb, int* __restrict__ cur,
                          int* __restrict__ sorted, int B) {
    int i = blockIdx.x * blockDim.x + threadIdx.x;
    if (i >= B) return;
    int pos = atomicAdd(&cur[tb[i]], 1);
    sorted[pos] = i;
}

// ---- pass 4a: precompute B operand in wave layout --------------------------
// B col g = [2c*p0, 2c*p1, -c, -c*|p|^2]; VGPR0 = K0|K2, VGPR1 = K1|K3
// bmat[(chunk*32 + lane)*2 + {0,1}]
__global__ void k_prep_b(const float* __restrict__ gp, float* __restrict__ bmat,
                         int G, int nChunk) {
    int i = blockIdx.x * blockDim.x + threadIdx.x;   // i = chunk*32 + lane
    if (i >= nChunk * 32) return;
    int lane = i & 31, c = i >> 5;
    int l16 = lane & 15, half = lane >> 4;
    int g = c * 16 + l16;
    int gc = g < G ? g : G - 1;                      // clamp: finite exp2 args
    float p0 = gp[2 * gc], p1 = gp[2 * gc + 1];
    bmat[2 * i]     = half ? (-C2LOG2E) : (2.0f * C2LOG2E * p0);
    bmat[2 * i + 1] = half ? (-C2LOG2E * (p0 * p0 + p1 * p1))
                           : (2.0f * C2LOG2E * p1);
}

// ---- pass 4b: precompute masked adjoints {m*a0, m*a1, m, 0}, zero-padded ---
__global__ void k_prep_adj(const float* __restrict__ adj, const int* __restrict__ gcnt,
                           float4* __restrict__ adjeff, int G, int Gpad, int NB) {
    int i = blockIdx.x * blockDim.x + threadIdx.x;   // i = bin*Gpad + g
    if (i >= NB * Gpad) return;
    int bin = i / Gpad, g = i - bin * Gpad;
    float4 v = make_float4(0.f, 0.f, 0.f, 0.f);
    if (g < G && gcnt[bin * G + g] > 0) {
        v.x = adj[((size_t)bin * G + g) * 2];
        v.y = adj[((size_t)bin * G + g) * 2 + 1];
        v.z = 1.0f;
    }
    adjeff[i] = v;
}

// ---- pass 5: WMMA compute --------------------------------------------------
// A row b = [x0, x1, |x|^2, 1]  (A: VGPR0 = K0|K2, VGPR1 = K1|K3; lane -> M)
// C/D: VGPR r -> row r (lanes 0-15) / r+8 (lanes 16-31); lane%16 -> col N
template <bool PRE>
__global__ __launch_bounds__(CTHREADS)
void k_compute(const float* __restrict__ x, const float* __restrict__ gp,
               const float* __restrict__ adj, const int* __restrict__ gcnt,
               const int* __restrict__ sorted, const int* __restrict__ offs,
               const float* __restrict__ bmat, const float4* __restrict__ adjeff,
               float* __restrict__ out, int G, int Gpad) {
    const int bin   = blockIdx.x;
    const int start = offs[bin];
    const int nbin  = offs[bin + 1] - start;
    const int wave  = threadIdx.x >> 5;
    const int lane  = threadIdx.x & 31;
    const int row0  = (blockIdx.y * WAVES_PER_BLOCK + wave) * 16;
    if (row0 >= nbin) return;                 // wave-uniform: EXEC stays full

    const int l16  = lane & 15;
    const int half = lane >> 4;

    // loop-invariant A operand (16 samples x K=4)
    int sr  = row0 + l16;
    int src = sr < nbin ? sr : nbin - 1;      // padded rows: clamped, not written
    int si  = sorted[start + src];
    float x0 = x[2 * si], x1 = x[2 * si + 1];
    v2f A;
    A.x = half ? (x0 * x0 + x1 * x1) : x0;    // K2 : K0
    A.y = half ? 1.0f : x1;                   // K3 : K1

    const float*  adjb = adj  + (size_t)bin * G * 2;
    const int*    cntb = gcnt + (size_t)bin * G;
    const float4* aeb  = adjeff + (size_t)bin * Gpad;
    const v2f*    bmv  = (const v2f*)bmat;

    float acc0[8], acc1[8], accS[8];
#pragma unroll
    for (int r = 0; r < 8; ++r) { acc0[r] = 0.f; acc1[r] = 0.f; accS[r] = 0.f; }

    const int nChunk = (G + 15) >> 4;
    for (int c = 0; c < nChunk; ++c) {
        v2f Bm;
        float a0, a1, os;
        if constexpr (PRE) {
            Bm = bmv[c * 32 + lane];                       // global_load_b64
            float4 a = aeb[c * 16 + l16];                  // global_load_b128
            a0 = a.x; a1 = a.y; os = a.z;
        } else {
            int g  = (c << 4) + l16;
            int gc = g < G ? g : G - 1;
            float p0 = gp[2 * gc], p1 = gp[2 * gc + 1];
            Bm.x = half ? (-C2LOG2E) : (2.0f * C2LOG2E * p0);
            Bm.y = half ? (-C2LOG2E * (p0 * p0 + p1 * p1)) : (2.0f * C2LOG2E * p1);
            // UNCONDITIONAL loads (gc always valid) + selects: no EXEC branching
            int   cnt = cntb[gc];
            float aa0 = adjb[2 * gc];
            float aa1 = adjb[2 * gc + 1];
            bool  m   = (g < G) && (cnt > 0);
            a0 = m ? aa0 : 0.0f;
            a1 = m ? aa1 : 0.0f;
            os = m ? 1.0f : 0.0f;
        }

        v8f cz = {};
        v8f W = __builtin_amdgcn_wmma_f32_16x16x4_f32(
            false, A, false, Bm, (short)0, cz, false, false);

#pragma unroll
        for (int r = 0; r < 8; ++r) {
            float w = fast_exp2(W[r]);        // v_exp_f32 on TRANS unit
            acc0[r] = fmaf(w, a0, acc0[r]);
            acc1[r] = fmaf(w, a1, acc1[r]);
            accS[r] = fmaf(w, os, accS[r]);
        }
    }

    // reduce the 16 grid-columns held across lanes of each half-wave
#pragma unroll
    for (int r = 0; r < 8; ++r) {
#pragma unroll
        for (int off = 1; off < 16; off <<= 1) {
            acc0[r] += __shfl_xor(acc0[r], off, 32);
            acc1[r] += __shfl_xor(acc1[r], off, 32);
            accS[r] += __shfl_xor(accS[r], off, 32);
        }
    }

    if (l16 == 0) {                            // lane 0 -> rows 0..7, lane 16 -> 8..15
#pragma unroll
        for (int r = 0; r < 8; ++r) {
            int row = half ? (r + 8) : r;
            int s2  = row0 + row;
            if (s2 < nbin) {
                int so = sorted[start + s2];
                float inv = -1.0f / (accS[r] + 1e-10f);
                out[2 * so]     = acc0[r] * inv;
                out[2 * so + 1] = acc1[r] * inv;
            }
        }
    }
}

// ---------------------------------------------------------------------------
extern "C" void kernel_launch(void* const* d_in, const int* in_sizes, int n_in,
                              void* d_out, int out_size, void* d_ws, size_t ws_size,
                              hipStream_t stream) {
    const float* t    = (const float*)d_in[0];   // (B,1) f32
    const float* x    = (const float*)d_in[1];   // (B,2) f32
    const float* gp   = (const float*)d_in[2];   // (G,2) f32
    const float* adj  = (const float*)d_in[3];   // (NB,G,2) f32
    const float* te   = (const float*)d_in[4];   // (NB+1,) f32
    const int*   gcnt = (const int*)d_in[5];     // (NB,G) i32
    float* out = (float*)d_out;                  // (B,2) f32

    const int B  = in_sizes[0];
    const int G  = in_sizes[2] / 2;
    const int NB = in_sizes[5] / G;
    const int nChunk = (G + 15) / 16;
    const int Gpad   = nChunk * 16;

    // ws: ints [0:20) counts | [32:53) offsets | [64:84) cursors
    //     [96:96+B) tb | [96+B:96+2B) sorted | then f32 tables (256B aligned)
    int* ws     = (int*)d_ws;
    int* cnts   = ws;
    int* offsb  = ws + 32;
    int* cur    = ws + 64;
    int* tb     = ws + 96;
    int* sorted = ws + 96 + B;

    size_t bOff   = (((size_t)(96 + 2 * B) * 4 + 255) / 256) * 256;
    size_t bBytes = (size_t)nChunk * 32 * 2 * sizeof(float);
    size_t aOff   = ((bOff + bBytes + 255) / 256) * 256;
    size_t aBytes = (size_t)NB * Gpad * sizeof(float4);
    const bool pre = (aOff + aBytes) <= ws_size;

    float*  bmat   = (float*)((char*)d_ws + bOff);
    float4* adjeff = (float4*)((char*)d_ws + aOff);

    k_init<<<1, 128, 0, stream>>>(ws);
    k_bin<<<(B + 255) / 256, 256, 0, stream>>>(t, te, tb, cnts, B, NB);
    k_scan<<<1, 32, 0, stream>>>(cnts, offsb, cur, NB);
    k_scatter<<<(B + 255) / 256, 256, 0, stream>>>(tb, cur, sorted, B);

    const int tilesMax = (B + 15) / 16;          // worst case: one bin holds all
    const int gy = (tilesMax + WAVES_PER_BLOCK - 1) / WAVES_PER_BLOCK;
    dim3 grid(NB, gy);

    if (pre) {
        k_prep_b<<<(nChunk * 32 + 255) / 256, 256, 0, stream>>>(gp, bmat, G, nChunk);
        k_prep_adj<<<(NB * Gpad + 255) / 256, 256, 0, stream>>>(adj, gcnt, adjeff,
                                                                G, Gpad, NB);
        k_compute<true><<<grid, CTHREADS, 0, stream>>>(x, gp, adj, gcnt, sorted,
                                                       offsb, bmat, adjeff, out,
                                                       G, Gpad);
    } else {
        k_compute<false><<<grid, CTHREADS, 0, stream>>>(x, gp, adj, gcnt, sorted,
                                                        offsb, bmat, adjeff, out,
                                                        G, Gpad);
    }
}